// SoftDTW_5145370821009
// MI455X (gfx1250) — compile-verified
//
#include <hip/hip_runtime.h>
#include <cstddef>

typedef __attribute__((ext_vector_type(2))) float v2f;
typedef __attribute__((ext_vector_type(8))) float v8f;

#define BATCH 16
#define SEQ   512
#define DIM   64
#define MAXD  10000.0f
#define INF_VAL 1e30f

// ---------------------------------------------------------------------------
// Phase 0: row squared-norms of X and Y -> workspace (16x512 each)
// ---------------------------------------------------------------------------
__global__ __launch_bounds__(256) void row_norms_kernel(
    const float* __restrict__ X, const float* __restrict__ Y,
    float* __restrict__ xn, float* __restrict__ yn) {
  int idx = blockIdx.x * 256 + threadIdx.x;          // 0 .. 2*16*512-1
  const int nrows = BATCH * SEQ;
  bool isX = idx < nrows;
  int r = isX ? idx : idx - nrows;
  const float* src = isX ? X : Y;
  const float4* row = (const float4*)(src + (size_t)r * DIM);
  float s = 0.f;
#pragma unroll
  for (int q = 0; q < DIM / 4; ++q) {
    float4 v = row[q];
    s += v.x * v.x + v.y * v.y + v.z * v.z + v.w * v.w;
  }
  (isX ? xn : yn)[r] = s;
}

// ---------------------------------------------------------------------------
// Phase 1: D[b,i,j] = min(xn[i] + yn[j] - 2*X_i.Y_j, 10000) via fp32 WMMA.
// One wave32 per 16x16 tile.  V_WMMA_F32_16X16X4_F32 layout:
//   A 16x4 : lanes 0-15 M=lane K={k0,k0+1}; lanes 16-31 M=lane-16 K={k0+2,k0+3}
//   B 4x16 : lanes 0-15 N=lane K={k0,k0+1}; lanes 16-31 N=lane-16 K={k0+2,k0+3}
//   C/D    : VGPR v, lanes 0-15 -> (M=v, N=lane), lanes 16-31 -> (M=v+8, N=lane-16)
// ---------------------------------------------------------------------------
__global__ __launch_bounds__(32) void dist_wmma_kernel(
    const float* __restrict__ X, const float* __restrict__ Y,
    const float* __restrict__ xn, const float* __restrict__ yn,
    float* __restrict__ D) {
  const int lane = threadIdx.x;
  const int b  = blockIdx.z;
  const int ti = blockIdx.y;
  const int tj = blockIdx.x;
  const int m  = lane & 15;
  const int hi = lane >> 4;

  const float* __restrict__ Xr = X + (((size_t)b * SEQ) + (size_t)ti * 16 + m) * DIM;
  const float* __restrict__ Yr = Y + (((size_t)b * SEQ) + (size_t)tj * 16 + m) * DIM;

  v8f acc = {};
#pragma unroll
  for (int k0 = 0; k0 < DIM; k0 += 4) {
    const int ka = k0 + 2 * hi;
    v2f a;  a.x  = Xr[ka]; a.y  = Xr[ka + 1];
    v2f bb; bb.x = Yr[ka]; bb.y = Yr[ka + 1];
    // D = A x B + C  (fp32 tensor op)
    acc = __builtin_amdgcn_wmma_f32_16x16x4_f32(
        /*neg_a=*/false, a, /*neg_b=*/false, bb,
        /*c_mod=*/(short)0, acc, /*reuse_a=*/false, /*reuse_b=*/false);
  }

  const float ynv = yn[(size_t)b * SEQ + (size_t)tj * 16 + m];
  float* __restrict__ Dtile =
      D + (((size_t)b * SEQ) + (size_t)ti * 16) * SEQ + (size_t)tj * 16 + m;
#pragma unroll
  for (int v = 0; v < 8; ++v) {
    const int M = v + (hi ? 8 : 0);
    float xv = xn[(size_t)b * SEQ + (size_t)ti * 16 + M];
    float d  = xv + ynv - 2.0f * acc[v];
    d = fminf(d, MAXD);
    Dtile[(size_t)M * SEQ] = d;
  }
}

// ---------------------------------------------------------------------------
// Phase 2: soft-DTW anti-diagonal wavefront. One workgroup per batch,
// thread t owns row i = t+1; 3 rotating diagonal buffers in LDS.
// D element for the next diagonal is prefetched before the LDS reads so the
// L2 access overlaps the exp/log math and the workgroup barrier.
// ---------------------------------------------------------------------------
__global__ __launch_bounds__(SEQ) void softdtw_dp_kernel(
    const float* __restrict__ D, float* __restrict__ out) {
  const int b = blockIdx.x;
  const int t = threadIdx.x;       // 0..511
  const int i = t + 1;             // DP row, 1..512
  __shared__ float buf[3][SEQ + 1];

  if (t == 0) {
    buf[0][0] = 0.0f;              // R[0][0]
    buf[1][0] = INF_VAL;           // R[0][1]
    buf[1][1] = INF_VAL;           // R[1][0]
  }
  __syncthreads();

  const float* __restrict__ Db = D + (size_t)b * SEQ * SEQ + (size_t)t * SEQ; // row i-1

  int j2 = 2 - i;
  float dcur = (j2 >= 1 && j2 <= SEQ) ? Db[j2 - 1] : 0.0f;

  for (int d = 2; d <= 2 * SEQ; ++d) {
    // prefetch D for diagonal d+1 (sequential within this thread's row)
    const int jn = d + 1 - i;
    float dnext = (jn >= 1 && jn <= SEQ) ? Db[jn - 1] : 0.0f;

    float* cur       = buf[d % 3];
    const float* p1  = buf[(d + 2) % 3];   // diagonal d-1
    const float* p2  = buf[(d + 1) % 3];   // diagonal d-2
    const int j = d - i;
    if (j >= 1 && j <= SEQ) {
      float r0 = p2[i - 1];                // R[i-1][j-1]
      float r1 = p1[i - 1];                // R[i-1][j]
      float r2 = p1[i];                    // R[i][j-1]
      float mn = fminf(fminf(r0, r1), r2);
      float s  = __expf(mn - r0) + __expf(mn - r1) + __expf(mn - r2);
      cur[i] = dcur + (mn - __logf(s));    // gamma = 1
    } else if (j == 0) {
      cur[i] = INF_VAL;                    // R[d][0]
    }
    if (t == 0 && d <= SEQ) cur[0] = INF_VAL;  // R[0][d]
    __syncthreads();
    dcur = dnext;
  }

  if (t == SEQ - 1) out[b] = buf[(2 * SEQ) % 3][SEQ];  // R[N][M]
}

// ---------------------------------------------------------------------------
extern "C" void kernel_launch(void* const* d_in, const int* in_sizes, int n_in,
                              void* d_out, int out_size, void* d_ws, size_t ws_size,
                              hipStream_t stream) {
  const float* X = (const float*)d_in[0];
  const float* Y = (const float*)d_in[1];
  float* out = (float*)d_out;

  float* D  = (float*)d_ws;                         // 16*512*512 f32 = 16 MB
  size_t dElems = (size_t)BATCH * SEQ * SEQ;
  float* xn = D + dElems;                           // 16*512 f32
  float* yn = xn + (size_t)BATCH * SEQ;             // 16*512 f32

  row_norms_kernel<<<dim3((2 * BATCH * SEQ) / 256), 256, 0, stream>>>(X, Y, xn, yn);
  dist_wmma_kernel<<<dim3(SEQ / 16, SEQ / 16, BATCH), 32, 0, stream>>>(X, Y, xn, yn, D);
  softdtw_dp_kernel<<<dim3(BATCH), SEQ, 0, stream>>>(D, out);
}